// CompGCNTransE_84645215469846
// MI455X (gfx1250) — compile-verified
//
#include <hip/hip_runtime.h>
#include <math.h>

#define N_NODES_C 20000
#define N_EDGES_C 320000
#define N_RELS_C  500
#define EMB_C     256
#define D4_C      64
#define BQ_C      32
#define GAMMA_C   12.0f
#define BN_EPS_C  1e-5f
#define LDS_STRIDE 68   // 64 + 4 pad dwords: distinct banks for B-fragment reads

typedef float v2f __attribute__((ext_vector_type(2)));
typedef float v8f __attribute__((ext_vector_type(8)));

#define WMMA_F32(a, b, c) \
  __builtin_amdgcn_wmma_f32_16x16x4_f32(false, (a), false, (b), (short)0, (c), false, false)

// A-fragment loader: A row-major [.. x lda], 16x4 tile at (row0, k).
// Lanes 0-15: M=lane, VGPR0=K=k+0, VGPR1=K=k+1; lanes 16-31: K=k+2,k+3.
__device__ __forceinline__ v2f ld_a(const float* A, int row0, int lda, int k, int lane) {
  int m  = lane & 15;
  int ks = (lane >> 4) << 1;
  const float* p = A + (size_t)(row0 + m) * lda + k + ks;
  v2f r; r.x = p[0]; r.y = p[1];
  return r;
}

// ---------------- degrees ----------------
__global__ void k_degree(const int* __restrict__ src, const int* __restrict__ dst,
                         float* deg_in, float* deg_out, const int* flag) {
  if (*flag == 0) return;
  int e = blockIdx.x * blockDim.x + threadIdx.x;
  if (e >= N_EDGES_C) return;
  atomicAdd(deg_in + dst[e], 1.0f);
  atomicAdd(deg_out + src[e], 1.0f);
}

// ---------------- layer-1 edge scatter (256-dim messages, float4 gathers) ----------------
__global__ void k_scatter1(const float* __restrict__ node, const float* __restrict__ eemb,
                           const int* __restrict__ src, const int* __restrict__ dst,
                           const int* __restrict__ evid,
                           float* ho1, float* hi1, const int* flag) {
  if (*flag == 0) return;
  int e  = blockIdx.x * 4 + (threadIdx.x >> 6);   // 4 edges per 256-thread block
  int d4 = threadIdx.x & 63;                      // float4 chunk within 256 dims
  int s = src[e], t = dst[e], r = evid[e];
  float4 ns = *(const float4*)(node + (size_t)s * EMB_C + d4 * 4);
  float4 nt = *(const float4*)(node + (size_t)t * EMB_C + d4 * 4);
  float4 ev = *(const float4*)(eemb + (size_t)r * EMB_C + d4 * 4);
  float* po = ho1 + (size_t)t * EMB_C + d4 * 4;
  float* pi = hi1 + (size_t)s * EMB_C + d4 * 4;
  atomicAdd(po + 0, ns.x - ev.x); atomicAdd(po + 1, ns.y - ev.y);
  atomicAdd(po + 2, ns.z - ev.z); atomicAdd(po + 3, ns.w - ev.w);
  atomicAdd(pi + 0, nt.x - ev.x); atomicAdd(pi + 1, nt.y - ev.y);
  atomicAdd(pi + 2, nt.z - ev.z); atomicAdd(pi + 3, nt.w - ev.w);
}

// ---------------- divide rows by clamped degree (float4) ----------------
__global__ void k_degnorm(float* a, const float* __restrict__ deg, int Ddiv4, int total4,
                          const int* flag) {
  if (*flag == 0) return;
  int i = blockIdx.x * blockDim.x + threadIdx.x;
  if (i >= total4) return;
  float inv = 1.0f / fmaxf(deg[i / Ddiv4], 1.0f);
  float4* p = (float4*)a + i;
  float4 v = *p;
  v.x *= inv; v.y *= inv; v.z *= inv; v.w *= inv;
  *p = v;
}

// ---------------- layer-1 fused triple GEMM + BN + tanh (WMMA f32, LDS-staged B) ----------
// h1[N,64] = tanh(BN((ho@WO^T + hi@WI^T + h@WS^T + biases)/3))
__global__ __launch_bounds__(256) void k_layer1(
    const float* __restrict__ ho, const float* __restrict__ hi, const float* __restrict__ h,
    const float* __restrict__ WO, const float* __restrict__ bO,
    const float* __restrict__ WI, const float* __restrict__ bI,
    const float* __restrict__ WS, const float* __restrict__ bS,
    const float* __restrict__ g, const float* __restrict__ bb,
    const float* __restrict__ m, const float* __restrict__ v,
    float* __restrict__ h1, const int* flag) {
  __shared__ float sw[3][64][LDS_STRIDE];   // 64-col K-panel of each weight matrix
  if (*flag == 0) return;                   // uniform: no barrier reached
  int tid  = threadIdx.x;
  int lane = tid & 31;
  int gw   = (blockIdx.x * blockDim.x + tid) >> 5;
  bool active = (gw < N_NODES_C / 16);
  int row0 = (active ? gw : N_NODES_C / 16 - 1) * 16;

  v8f acc[4];
#pragma unroll
  for (int nt = 0; nt < 4; ++nt)
#pragma unroll
    for (int r = 0; r < 8; ++r) acc[nt][r] = 0.0f;

  const float* Wmats[3] = {WO, WI, WS};
  int nidx = lane & 15;
  int ks   = (lane >> 4) << 1;

  for (int kp = 0; kp < 4; ++kp) {          // 4 panels of K=64
    __syncthreads();                         // protect LDS from previous panel's readers
#pragma unroll
    for (int mat = 0; mat < 3; ++mat) {
      const float* W = Wmats[mat];
      for (int f = tid; f < 1024; f += 256) {   // 64 rows x 16 float4
        int rowp = f >> 4, kc4 = f & 15;
        float4 w4 = *(const float4*)(W + (size_t)rowp * EMB_C + kp * 64 + kc4 * 4);
        *(float4*)(&sw[mat][rowp][kc4 * 4]) = w4;
      }
    }
    __syncthreads();
    for (int kk = 0; kk < 64; kk += 4) {
      int k = kp * 64 + kk;
      v2f a0 = ld_a(ho, row0, EMB_C, k, lane);
      v2f a1 = ld_a(hi, row0, EMB_C, k, lane);
      v2f a2 = ld_a(h,  row0, EMB_C, k, lane);
#pragma unroll
      for (int nt = 0; nt < 4; ++nt) {
        acc[nt] = WMMA_F32(a0, *(const v2f*)(&sw[0][nt * 16 + nidx][kk + ks]), acc[nt]);
        acc[nt] = WMMA_F32(a1, *(const v2f*)(&sw[1][nt * 16 + nidx][kk + ks]), acc[nt]);
        acc[nt] = WMMA_F32(a2, *(const v2f*)(&sw[2][nt * 16 + nidx][kk + ks]), acc[nt]);
      }
    }
  }
  if (!active) return;

  int cl = lane & 15;
  int ro = (lane >> 4) * 8;
#pragma unroll
  for (int nt = 0; nt < 4; ++nt) {
    int n = nt * 16 + cl;
    float bias = bO[n] + bI[n] + bS[n];
    float inv  = rsqrtf(v[n] + BN_EPS_C);
    float gg = g[n], bv = bb[n], mm = m[n];
#pragma unroll
    for (int r = 0; r < 8; ++r) {
      float val = (acc[nt][r] + bias) * (1.0f / 3.0f);
      val = (val - mm) * inv * gg + bv;
      h1[(size_t)(row0 + ro + r) * D4_C + n] = tanhf(val);
    }
  }
}

// ---------------- per-relation transform: t = eemb@Wr1^T+br1 ; rt = tanh(t) --------------
__global__ void k_rel1(const float* __restrict__ eemb, const float* __restrict__ Wr1,
                       const float* __restrict__ br1, float* t, float* rt) {
  int i = blockIdx.x * blockDim.x + threadIdx.x;
  if (i >= N_RELS_C * D4_C) return;
  int r = i / D4_C, j = i % D4_C;
  const float4* ep = (const float4*)(eemb + (size_t)r * EMB_C);
  const float4* wp = (const float4*)(Wr1 + (size_t)j * EMB_C);
  float s = br1[j];
  for (int k = 0; k < EMB_C / 4; ++k) {
    float4 a = ep[k], b = wp[k];
    s += a.x * b.x + a.y * b.y + a.z * b.z + a.w * b.w;
  }
  t[i]  = s;
  rt[i] = tanhf(s);
}

// ---------------- layer-2 edge scatter (64-dim messages, float4 gathers) ----------------
__global__ void k_scatter2(const float* __restrict__ h1, const float* __restrict__ rt,
                           const int* __restrict__ src, const int* __restrict__ dst,
                           const int* __restrict__ evid,
                           float* ho2, float* hi2, const int* flag) {
  if (*flag == 0) return;
  int e  = blockIdx.x * 16 + (threadIdx.x >> 4);  // 16 edges per 256-thread block
  int d4 = threadIdx.x & 15;                      // float4 chunk within 64 dims
  int s = src[e], t = dst[e], r = evid[e];
  float4 hs = *(const float4*)(h1 + (size_t)s * D4_C + d4 * 4);
  float4 ht = *(const float4*)(h1 + (size_t)t * D4_C + d4 * 4);
  float4 ev = *(const float4*)(rt + (size_t)r * D4_C + d4 * 4);
  float* po = ho2 + (size_t)t * D4_C + d4 * 4;
  float* pi = hi2 + (size_t)s * D4_C + d4 * 4;
  atomicAdd(po + 0, hs.x - ev.x); atomicAdd(po + 1, hs.y - ev.y);
  atomicAdd(po + 2, hs.z - ev.z); atomicAdd(po + 3, hs.w - ev.w);
  atomicAdd(pi + 0, ht.x - ev.x); atomicAdd(pi + 1, ht.y - ev.y);
  atomicAdd(pi + 2, ht.z - ev.z); atomicAdd(pi + 3, ht.w - ev.w);
}

// ---------------- layer-2 fused triple GEMM + BN + tanh (WMMA f32, LDS-staged B) ----------
// grid = (157, 4): blockIdx.y selects the 64-wide output column group; each block stages
// its three 64x64 weight slices once. hn[N,256]; when flag==0 copies node_embs instead.
__global__ __launch_bounds__(256) void k_layer2(
    const float* __restrict__ ho, const float* __restrict__ hi, const float* __restrict__ h1,
    const float* __restrict__ WO, const float* __restrict__ bO,
    const float* __restrict__ WI, const float* __restrict__ bI,
    const float* __restrict__ WS, const float* __restrict__ bS,
    const float* __restrict__ g, const float* __restrict__ bb,
    const float* __restrict__ m, const float* __restrict__ v,
    const float* __restrict__ node, float* __restrict__ hn, const int* flag) {
  __shared__ float sw[3][64][LDS_STRIDE];
  int tid  = threadIdx.x;
  int lane = tid & 31;
  int mt   = blockIdx.x * 8 + (tid >> 5);
  int cg   = blockIdx.y;
  bool active = (mt < N_NODES_C / 16);
  int row0 = (active ? mt : N_NODES_C / 16 - 1) * 16;
  int c0   = cg * 64;

  if (*flag == 0) {                 // uniform branch: pass node_embs through
    if (!active) return;
    for (int r = 0; r < 16; ++r)
      for (int c = lane; c < 64; c += 32) {
        size_t off = (size_t)(row0 + r) * EMB_C + c0 + c;
        hn[off] = node[off];
      }
    return;
  }

  const float* Wmats[3] = {WO, WI, WS};
#pragma unroll
  for (int mat = 0; mat < 3; ++mat) {
    const float* W = Wmats[mat];
    for (int f = tid; f < 1024; f += 256) {   // rows c0..c0+63, all K=64
      int rowp = f >> 4, kc4 = f & 15;
      float4 w4 = *(const float4*)(W + (size_t)(c0 + rowp) * D4_C + kc4 * 4);
      *(float4*)(&sw[mat][rowp][kc4 * 4]) = w4;
    }
  }
  __syncthreads();

  v8f acc[4];
#pragma unroll
  for (int nt = 0; nt < 4; ++nt)
#pragma unroll
    for (int r = 0; r < 8; ++r) acc[nt][r] = 0.0f;

  int nidx = lane & 15;
  int ks   = (lane >> 4) << 1;
  for (int k = 0; k < D4_C; k += 4) {
    v2f a0 = ld_a(ho, row0, D4_C, k, lane);
    v2f a1 = ld_a(hi, row0, D4_C, k, lane);
    v2f a2 = ld_a(h1, row0, D4_C, k, lane);
#pragma unroll
    for (int nt = 0; nt < 4; ++nt) {
      acc[nt] = WMMA_F32(a0, *(const v2f*)(&sw[0][nt * 16 + nidx][k + ks]), acc[nt]);
      acc[nt] = WMMA_F32(a1, *(const v2f*)(&sw[1][nt * 16 + nidx][k + ks]), acc[nt]);
      acc[nt] = WMMA_F32(a2, *(const v2f*)(&sw[2][nt * 16 + nidx][k + ks]), acc[nt]);
    }
  }
  if (!active) return;

  int cl = lane & 15;
  int ro = (lane >> 4) * 8;
#pragma unroll
  for (int nt = 0; nt < 4; ++nt) {
    int n = c0 + nt * 16 + cl;
    float bias = bO[n] + bI[n] + bS[n];
    float inv  = rsqrtf(v[n] + BN_EPS_C);
    float gg = g[n], bv = bb[n], mm = m[n];
#pragma unroll
    for (int r = 0; r < 8; ++r) {
      float val = (acc[nt][r] + bias) * (1.0f / 3.0f);
      val = (val - mm) * inv * gg + bv;
      hn[(size_t)(row0 + ro + r) * EMB_C + n] = tanhf(val);
    }
  }
}

// ---------------- query vectors: q = heads+rels (tail pred) or tails-rels (head pred) ----
__global__ void k_relsq(const float* __restrict__ t, const float* __restrict__ Wr2,
                        const float* __restrict__ br2,
                        const int* __restrict__ hids, const int* __restrict__ rids,
                        const int* __restrict__ tids,
                        const float* __restrict__ hn, const int* head_pred,
                        float* __restrict__ q) {
  int i = blockIdx.x;     // 0..31
  int d = threadIdx.x;    // 0..255
  const float* tp = t + (size_t)rids[i] * D4_C;
  const float* wp = Wr2 + (size_t)d * D4_C;
  float s = br2[d];
  for (int k = 0; k < D4_C; ++k) s += fmaxf(tp[k], 0.0f) * wp[k];
  float qv;
  if (*head_pred == 1) qv = hn[(size_t)tids[i] * EMB_C + d] - s;
  else                 qv = hn[(size_t)hids[i] * EMB_C + d] + s;
  q[(size_t)i * EMB_C + d] = qv;
}

// ---------------- scoring: dist(i, j) depends only on (i, j//32) ----------------
__global__ void k_score(const float* __restrict__ q, const float* __restrict__ hn,
                        float* __restrict__ score) {
  int gw   = (blockIdx.x * blockDim.x + threadIdx.x) >> 5;
  int lane = threadIdx.x & 31;
  if (gw >= BQ_C * 625) return;
  int i = gw / 625, kk = gw % 625;
  const float4* qp = (const float4*)(q + (size_t)i * EMB_C);
  const float4* hp = (const float4*)(hn + (size_t)(i * 625 + kk) * EMB_C);
  float s = 0.0f;
#pragma unroll
  for (int d = lane; d < EMB_C / 4; d += 32) {
    float4 a = qp[d], b = hp[d];
    float dx = a.x - b.x, dy = a.y - b.y, dz = a.z - b.z, dw = a.w - b.w;
    s += dx * dx + dy * dy + dz * dz + dw * dw;
  }
#pragma unroll
  for (int off = 16; off > 0; off >>= 1) s += __shfl_xor(s, off, 32);
  float sc = 1.0f / (1.0f + expf(sqrtf(s) - GAMMA_C));
  score[(size_t)i * N_NODES_C + kk * 32 + lane] = sc;
}

extern "C" void kernel_launch(void* const* d_in, const int* in_sizes, int n_in,
                              void* d_out, int out_size, void* d_ws, size_t ws_size,
                              hipStream_t stream) {
  const float* node = (const float*)d_in[0];
  const float* eemb = (const float*)d_in[1];
  const float* W_O1 = (const float*)d_in[2];  const float* b_O1 = (const float*)d_in[3];
  const float* W_I1 = (const float*)d_in[4];  const float* b_I1 = (const float*)d_in[5];
  const float* W_S1 = (const float*)d_in[6];  const float* b_S1 = (const float*)d_in[7];
  const float* bn1g = (const float*)d_in[8];  const float* bn1b = (const float*)d_in[9];
  const float* bn1m = (const float*)d_in[10]; const float* bn1v = (const float*)d_in[11];
  const float* Wr1  = (const float*)d_in[12]; const float* br1  = (const float*)d_in[13];
  const float* W_O2 = (const float*)d_in[14]; const float* b_O2 = (const float*)d_in[15];
  const float* W_I2 = (const float*)d_in[16]; const float* b_I2 = (const float*)d_in[17];
  const float* W_S2 = (const float*)d_in[18]; const float* b_S2 = (const float*)d_in[19];
  const float* bn2g = (const float*)d_in[20]; const float* bn2b = (const float*)d_in[21];
  const float* bn2m = (const float*)d_in[22]; const float* bn2v = (const float*)d_in[23];
  const float* Wr2  = (const float*)d_in[24]; const float* br2  = (const float*)d_in[25];
  const int* src  = (const int*)d_in[26];
  const int* dst  = (const int*)d_in[27];
  const int* evid = (const int*)d_in[28];
  const int* hids = (const int*)d_in[29];
  const int* rids = (const int*)d_in[30];
  const int* tids = (const int*)d_in[31];
  const int* fupd = (const int*)d_in[32];
  const int* fhead = (const int*)d_in[33];

  // workspace layout (floats)
  float* ws      = (float*)d_ws;
  float* deg_in  = ws;
  float* deg_out = deg_in + N_NODES_C;
  float* ho1 = deg_out + N_NODES_C;                  // [N, 256]
  float* hi1 = ho1 + (size_t)N_NODES_C * EMB_C;      // [N, 256]
  float* ho2 = hi1 + (size_t)N_NODES_C * EMB_C;      // [N, 64]
  float* hi2 = ho2 + (size_t)N_NODES_C * D4_C;       // [N, 64]
  float* h1  = hi2 + (size_t)N_NODES_C * D4_C;       // [N, 64]
  float* t500  = h1 + (size_t)N_NODES_C * D4_C;      // [500, 64]
  float* rt500 = t500 + (size_t)N_RELS_C * D4_C;     // [500, 64]
  float* q     = rt500 + (size_t)N_RELS_C * D4_C;    // [32, 256]

  size_t zbytes = (size_t)(2 * N_NODES_C + 2 * (size_t)N_NODES_C * EMB_C +
                           2 * (size_t)N_NODES_C * D4_C) * sizeof(float);
  hipMemsetAsync(d_ws, 0, zbytes, stream);

  float* hn    = (float*)d_out;                      // [N, 256]
  float* score = hn + (size_t)N_NODES_C * EMB_C;     // [32, N]

  k_degree<<<(N_EDGES_C + 255) / 256, 256, 0, stream>>>(src, dst, deg_in, deg_out, fupd);
  k_scatter1<<<N_EDGES_C / 4, 256, 0, stream>>>(node, eemb, src, dst, evid, ho1, hi1, fupd);
  k_degnorm<<<(N_NODES_C * EMB_C / 4 + 255) / 256, 256, 0, stream>>>(
      ho1, deg_in, EMB_C / 4, N_NODES_C * EMB_C / 4, fupd);
  k_degnorm<<<(N_NODES_C * EMB_C / 4 + 255) / 256, 256, 0, stream>>>(
      hi1, deg_out, EMB_C / 4, N_NODES_C * EMB_C / 4, fupd);
  k_layer1<<<((N_NODES_C / 16) * 32 + 255) / 256, 256, 0, stream>>>(
      ho1, hi1, node, W_O1, b_O1, W_I1, b_I1, W_S1, b_S1, bn1g, bn1b, bn1m, bn1v, h1, fupd);
  k_rel1<<<(N_RELS_C * D4_C + 255) / 256, 256, 0, stream>>>(eemb, Wr1, br1, t500, rt500);
  k_scatter2<<<N_EDGES_C / 16, 256, 0, stream>>>(h1, rt500, src, dst, evid, ho2, hi2, fupd);
  k_degnorm<<<(N_NODES_C * D4_C / 4 + 255) / 256, 256, 0, stream>>>(
      ho2, deg_in, D4_C / 4, N_NODES_C * D4_C / 4, fupd);
  k_degnorm<<<(N_NODES_C * D4_C / 4 + 255) / 256, 256, 0, stream>>>(
      hi2, deg_out, D4_C / 4, N_NODES_C * D4_C / 4, fupd);
  {
    dim3 grid2((N_NODES_C / 16 + 7) / 8, 4);
    k_layer2<<<grid2, 256, 0, stream>>>(
        ho2, hi2, h1, W_O2, b_O2, W_I2, b_I2, W_S2, b_S2, bn2g, bn2b, bn2m, bn2v,
        node, hn, fupd);
  }
  k_relsq<<<BQ_C, EMB_C, 0, stream>>>(t500, Wr2, br2, hids, rids, tids, hn, fhead, q);
  k_score<<<(BQ_C * 625 * 32 + 255) / 256, 256, 0, stream>>>(q, hn, score);
}